// BP_Head_Phase_2989297238616
// MI455X (gfx1250) — compile-verified
//
#include <hip/hip_runtime.h>
#include <math.h>

typedef float v2f __attribute__((ext_vector_type(2)));
typedef float v8f __attribute__((ext_vector_type(8)));

#define B_   1024
#define T_   500
#define KP   512      // padded time / freq dimension
#define HID  16

// workspace layout (in floats)
#define OFF_X     0
#define OFF_COS   (OFF_X   + B_*KP)     // 524288
#define OFF_SIN   (OFF_COS + KP*KP)     // +262144
#define OFF_PH    (OFF_SIN + KP*KP)     // +262144
#define OFF_WEFF  (OFF_PH  + B_*KP)     // +524288
#define OFF_BEFF  (OFF_WEFF + KP*2)

// ---------------------------------------------------------------------------
// 1) Spatial mean over the full 7x7 window.  100 MB streamed once from HBM.
//    Writes zero-padded x (1024 x 512).
// ---------------------------------------------------------------------------
__global__ void k_pool(const float* __restrict__ in, float* __restrict__ xb) {
  int idx = blockIdx.x * 256 + threadIdx.x;     // 0 .. B_*KP
  int b = idx >> 9, t = idx & (KP - 1);
  float s = 0.f;
  if (t < T_) {
    const float* p = in + ((long)b * T_ + t) * 49;
    #pragma unroll
    for (int i = 0; i < 49; ++i) s += p[i];
    s *= (1.0f / 49.0f);
  }
  xb[idx] = s;
}

// ---------------------------------------------------------------------------
// 2) DFT twiddle tables: C[t,k]=cos(2*pi*t*k/500), S[t,k]=-sin(...).
//    Argument reduced mod 500 so sincosf stays accurate.  Zero padded.
// ---------------------------------------------------------------------------
__global__ void k_twiddle(float* __restrict__ cg, float* __restrict__ sg) {
  int idx = blockIdx.x * 256 + threadIdx.x;     // over KP*KP
  int t = idx >> 9, k = idx & (KP - 1);
  float c = 0.f, s = 0.f;
  if (t < T_ && k < T_) {
    int m = (t * k) % T_;
    float th = (float)m * (6.283185307179586f / (float)T_);
    float sv, cv;
    sincosf(th, &sv, &cv);
    c = cv;
    s = -sv;   // FFT uses exp(-i*theta)
  }
  cg[idx] = c;
  sg[idx] = s;
}

// ---------------------------------------------------------------------------
// 3) Collapse the two Linears (dropout = identity):
//    Weff = W1 @ W2  (500x2, zero padded to 512x2),  beff = b1 @ W2 + b2.
// ---------------------------------------------------------------------------
__global__ void k_weff(const float* __restrict__ W1, const float* __restrict__ b1,
                       const float* __restrict__ W2, const float* __restrict__ b2,
                       float* __restrict__ weff, float* __restrict__ beff) {
  int idx = blockIdx.x * 256 + threadIdx.x;     // 0..511
  float w0 = 0.f, w1 = 0.f;
  if (idx < T_) {
    #pragma unroll
    for (int j = 0; j < HID; ++j) {
      float w = W1[idx * HID + j];
      w0 += w * W2[2 * j];
      w1 += w * W2[2 * j + 1];
    }
  }
  weff[2 * idx]     = w0;
  weff[2 * idx + 1] = w1;
  if (idx < 2) {
    float s = b2[idx];
    for (int j = 0; j < HID; ++j) s += b1[j] * W2[2 * j + idx];
    beff[idx] = s;
  }
}

// ---------------------------------------------------------------------------
// 4) DFT-as-GEMM + phase, using V_WMMA_F32_16X16X4_F32 (fp32 WMMA).
//    Block = 256 threads = 8 waves arranged 2(M) x 4(N); block tile 32M x 64N;
//    K staged through LDS in BK=32 slabs (strides padded vs bank conflicts).
//    Each wave keeps Re and Im 16x16 accumulators and issues 2 WMMAs / K-step.
//    Epilogue: phase = atan2f(Im, Re).
// ---------------------------------------------------------------------------
__global__ __launch_bounds__(256)
void k_dft_phase(const float* __restrict__ xb, const float* __restrict__ cg,
                 const float* __restrict__ sg, float* __restrict__ ph) {
  __shared__ float lA[32][34];   // x tile   [m][k]
  __shared__ float lC[32][66];   // cos tile [k][n]
  __shared__ float lS[32][66];   // sin tile [k][n]

  const int tid   = threadIdx.x;
  const int lane  = tid & 31;
  const int wid   = tid >> 5;
  const int waveM = wid >> 2;          // 0..1
  const int waveN = wid & 3;           // 0..3
  const int m_block = blockIdx.y * 32;
  const int n_block = blockIdx.x * 64;

  v8f accRe = {};
  v8f accIm = {};

  // ISA fragment layout for 32-bit 16x4 A / 4x16 B (wave32):
  // lanes 0-15 hold K={0,1}, lanes 16-31 hold K={2,3}; M/N = lane & 15.
  const int row  = lane & 15;
  const int koff = (lane >> 4) << 1;

  for (int k0 = 0; k0 < KP; k0 += 32) {
    __syncthreads();
    #pragma unroll
    for (int i = 0; i < 4; ++i) {              // 32x32 A tile
      int e = tid + i * 256;
      int r = e >> 5, c = e & 31;
      lA[r][c] = xb[(m_block + r) * KP + k0 + c];
    }
    #pragma unroll
    for (int i = 0; i < 8; ++i) {              // 32x64 B tiles (cos & sin)
      int e = tid + i * 256;
      int r = e >> 6, c = e & 63;
      lC[r][c] = cg[(k0 + r) * KP + n_block + c];
      lS[r][c] = sg[(k0 + r) * KP + n_block + c];
    }
    __syncthreads();

    #pragma unroll
    for (int kk = 0; kk < 32; kk += 4) {
      v2f a, bc, bs;
      a.x  = lA[waveM * 16 + row][kk + koff];
      a.y  = lA[waveM * 16 + row][kk + koff + 1];
      bc.x = lC[kk + koff    ][waveN * 16 + row];
      bc.y = lC[kk + koff + 1][waveN * 16 + row];
      bs.x = lS[kk + koff    ][waveN * 16 + row];
      bs.y = lS[kk + koff + 1][waveN * 16 + row];
      accRe = __builtin_amdgcn_wmma_f32_16x16x4_f32(
          false, a, false, bc, (short)0, accRe, false, false);
      accIm = __builtin_amdgcn_wmma_f32_16x16x4_f32(
          false, a, false, bs, (short)0, accIm, false, false);
    }
  }

  // C/D layout: VGPR r -> M = r + 8*(lane>=16); N = lane & 15.
  const int mhalf = (lane >> 4) * 8;
  #pragma unroll
  for (int r = 0; r < 8; ++r) {
    int m = m_block + waveM * 16 + mhalf + r;
    int n = n_block + waveN * 16 + (lane & 15);
    ph[m * KP + n] = atan2f(accIm[r], accRe[r]);
  }
}

// ---------------------------------------------------------------------------
// 5) Final collapsed linear: out[b,:] = phase[b,:] @ Weff + beff.
//    One wave per batch row, shuffle reduction (wave32).
// ---------------------------------------------------------------------------
__global__ void k_mlp(const float* __restrict__ ph, const float* __restrict__ weff,
                      const float* __restrict__ beff, float* __restrict__ out) {
  int wid  = threadIdx.x >> 5;
  int lane = threadIdx.x & 31;
  int b = blockIdx.x * 8 + wid;
  float s0 = 0.f, s1 = 0.f;
  for (int k = lane; k < KP; k += 32) {        // padded tail contributes 0
    float p = ph[b * KP + k];
    s0 += p * weff[2 * k];
    s1 += p * weff[2 * k + 1];
  }
  #pragma unroll
  for (int off = 16; off > 0; off >>= 1) {
    s0 += __shfl_down(s0, off, 32);
    s1 += __shfl_down(s1, off, 32);
  }
  if (lane == 0) {
    out[2 * b]     = s0 + beff[0];
    out[2 * b + 1] = s1 + beff[1];
  }
}

// ---------------------------------------------------------------------------
extern "C" void kernel_launch(void* const* d_in, const int* in_sizes, int n_in,
                              void* d_out, int out_size, void* d_ws, size_t ws_size,
                              hipStream_t stream) {
  (void)in_sizes; (void)n_in; (void)out_size; (void)ws_size;
  const float* emb = (const float*)d_in[0];
  // d_in[1] = rBr : unused by the reference math
  const float* W1 = (const float*)d_in[2];
  const float* b1 = (const float*)d_in[3];
  const float* W2 = (const float*)d_in[4];
  const float* b2 = (const float*)d_in[5];

  float* ws   = (float*)d_ws;
  float* xb   = ws + OFF_X;
  float* cg   = ws + OFF_COS;
  float* sg   = ws + OFF_SIN;
  float* ph   = ws + OFF_PH;
  float* weff = ws + OFF_WEFF;
  float* beff = ws + OFF_BEFF;
  float* out  = (float*)d_out;

  k_pool   <<<(B_ * KP) / 256, 256, 0, stream>>>(emb, xb);
  k_twiddle<<<(KP * KP) / 256, 256, 0, stream>>>(cg, sg);
  k_weff   <<<2, 256, 0, stream>>>(W1, b1, W2, b2, weff, beff);
  k_dft_phase<<<dim3(KP / 64, B_ / 32), 256, 0, stream>>>(xb, cg, sg, ph);
  k_mlp    <<<B_ / 8, 256, 0, stream>>>(ph, weff, beff, out);
}